// StripConv_54563264528988
// MI455X (gfx1250) — compile-verified
//
#include <hip/hip_runtime.h>
#include <math.h>

// Problem constants (from reference setup_inputs)
#define B_  4
#define C_  64
#define H_  128
#define W_  128
#define K_  9
#define O_  64
#define HW_ (H_*W_)
#define EPSV 1e-5f

typedef float v2f __attribute__((ext_vector_type(2)));
typedef float v8f __attribute__((ext_vector_type(8)));

// ---------------------------------------------------------------------------
// Tiled conv RxR (pad R/2): one block per (b,h) row, computes ALL K_ output
// channels. x rows staged in LDS once per input channel; weights read with
// block-uniform indices -> scalar (s_load) fetches through K$.
// ---------------------------------------------------------------------------
template <int R>
__global__ __launch_bounds__(128) void conv_k(const float* __restrict__ x,
                                              const float* __restrict__ w,
                                              const float* __restrict__ bias,
                                              float* __restrict__ out) {
    constexpr int PAD = R / 2;
    constexpr int WH  = W_ + 2 * PAD;          // 130 (R=3) / 132 (R=5)
    __shared__ float xs[R * WH];

    int tid = threadIdx.x;                     // == pixel w (0..127)
    int bh  = blockIdx.x;
    int b   = bh / H_;
    int h   = bh % H_;

    float acc[K_];
    #pragma unroll
    for (int kc = 0; kc < K_; ++kc) acc[kc] = bias[kc];

    const float* xb = x + (long)b * C_ * HW_;

    for (int c = 0; c < C_; ++c) {
        __syncthreads();                       // xs reuse guard
        for (int i = tid; i < R * WH; i += 128) {
            int r  = i / WH, col = i % WH;
            int yy = h + r - PAD, xx = col - PAD;
            float v = 0.f;
            if (yy >= 0 && yy < H_ && xx >= 0 && xx < W_)
                v = xb[c * HW_ + yy * W_ + xx];
            xs[i] = v;
        }
        __syncthreads();

        float xv[R * R];
        #pragma unroll
        for (int dy = 0; dy < R; ++dy)
            #pragma unroll
            for (int dx = 0; dx < R; ++dx)
                xv[dy * R + dx] = xs[dy * WH + tid + dx];

        #pragma unroll
        for (int kc = 0; kc < K_; ++kc) {
            const float* wp = w + ((long)(kc * C_ + c)) * (R * R);  // uniform
            float a = acc[kc];
            #pragma unroll
            for (int j = 0; j < R * R; ++j) a = fmaf(xv[j], wp[j], a);
            acc[kc] = a;
        }
    }
    #pragma unroll
    for (int kc = 0; kc < K_; ++kc)
        out[((long)(b * K_ + kc)) * HW_ + h * W_ + tid] = acc[kc];
}

// ---------------------------------------------------------------------------
// GroupNorm stats: one block per (b, group). data layout [b][ch][HW].
// stats[bg*2] = mean ; stats[bg*2+1] = rsqrt(var + eps)
// ---------------------------------------------------------------------------
__global__ void gn_stats_k(const float* __restrict__ data, float* __restrict__ stats,
                           int chPerGroup, int groups) {
    __shared__ float ssum[256];
    __shared__ float ssum2[256];
    int bg = blockIdx.x;
    long base = (long)bg * chPerGroup * HW_;
    int n = chPerGroup * HW_;
    float s = 0.f, s2 = 0.f;
    for (int i = threadIdx.x; i < n; i += 256) {
        float v = data[base + i];
        s += v; s2 += v * v;
    }
    ssum[threadIdx.x] = s; ssum2[threadIdx.x] = s2;
    __syncthreads();
    for (int off = 128; off > 0; off >>= 1) {
        if (threadIdx.x < off) {
            ssum[threadIdx.x]  += ssum[threadIdx.x + off];
            ssum2[threadIdx.x] += ssum2[threadIdx.x + off];
        }
        __syncthreads();
    }
    if (threadIdx.x == 0) {
        float mu  = ssum[0] / (float)n;
        float var = ssum2[0] / (float)n - mu * mu;
        stats[bg * 2]     = mu;
        stats[bg * 2 + 1] = rsqrtf(var + EPSV);
    }
}

// ---------------------------------------------------------------------------
// Apply GN (+activation) in place. mode 0: tanh (offset branch),
// mode 1: identity (theta branch; tan/sin/cos(atan(v)) folded algebraically).
// ---------------------------------------------------------------------------
__global__ void gn_apply_k(float* __restrict__ data, const float* __restrict__ stats,
                           const float* __restrict__ gamma, const float* __restrict__ beta,
                           int Cc, int groups, int mode) {
    int idx = blockIdx.x * blockDim.x + threadIdx.x;
    if (idx >= B_ * Cc * HW_) return;
    int ch = (idx / HW_) % Cc;
    int b  = idx / (Cc * HW_);
    int g  = ch / (Cc / groups);
    float mu = stats[(b * groups + g) * 2];
    float rs = stats[(b * groups + g) * 2 + 1];
    float v = (data[idx] - mu) * rs * gamma[ch] + beta[ch];
    data[idx] = (mode == 0) ? tanhf(v) : v;
}

// ---------------------------------------------------------------------------
// Fused bilinear-sample + strip einsum, WMMA f32 16x16x4.
// One workgroup (256 thr = 8 waves) per (b, h) row of 128 pixels.
// LDS holds channel-PAIR-interleaved data so every A/B fragment is a single
// aligned ds_load_b64 straight into an even VGPR pair (no re-pairing movs).
// Row strides (in float2 units) are ==16 (mod 32) so the two lane-halves
// (rows cpair and cpair+1) hit disjoint 32-bank sets.
// ---------------------------------------------------------------------------
#define SR2 144   // slds2 row stride (float2 units), >=128, ==16 mod 32
#define WR2 80    // wlds2 row stride (float2 units), >=64,  ==16 mod 32

__global__ __launch_bounds__(256) void strip_wmma_k(
    const float* __restrict__ x, const float* __restrict__ off,
    const float* __restrict__ th, const float* __restrict__ wstrip,
    const float* __restrict__ bstrip, float* __restrict__ outraw) {

    __shared__ v2f slds2[32 * SR2];   // samples  [cpair][pixel]   (36 KB)
    __shared__ v2f wlds2[32 * WR2];   // weights  [cpair][o]       (20 KB)
    __shared__ int   iy0[W_], ix0[W_];
    __shared__ float cfy[W_], cfx[W_];

    int tid  = threadIdx.x;
    int bh   = blockIdx.x;
    int b    = bh / H_;
    int h    = bh % H_;
    int wv   = tid >> 5;        // wave id 0..7  -> pixel tile
    int lane = tid & 31;
    int half = lane >> 4;       // lane half selects K pair {0,1} vs {2,3}
    int l16  = lane & 15;

    v8f acc[4];
    #pragma unroll
    for (int mt = 0; mt < 4; ++mt)
        #pragma unroll
        for (int j = 0; j < 8; ++j) acc[mt][j] = 0.f;

    const float* xb = x + (long)b * C_ * HW_;

    for (int k = 0; k < K_; ++k) {
        float s = (float)(k - 4);
        int   m = 4 + ((k < 4) ? (4 - k) : (k - 4));   // mirrored tap index

        // --- per-pixel sampling coordinates (threads 0..127) ---
        if (tid < W_) {
            int wp = tid;
            float yn, xn;
            if (k == 4) {
                yn = (float)h; xn = (float)wp;          // center tap: identity
            } else {
                long base = ((long)(b * K_ + m)) * HW_ + h * W_ + wp;
                float v = th[base];                      // GN'd theta value
                float o = off[base];                     // tanh'd offset
                float inv = rsqrtf(1.f + v * v);         // cos(atan(v))
                float yoff = s * v + o * inv;            // s*tan + off*cos
                float xoff = s - (v * inv) * o;          // s - sin*off
                yn = (float)h  + s + yoff;
                xn = (float)wp + s + xoff;
            }
            yn = fminf(fmaxf(yn, 0.f), (float)(H_ - 1));
            xn = fminf(fmaxf(xn, 0.f), (float)(W_ - 1));
            float y0 = floorf(yn), x0 = floorf(xn);
            iy0[wp] = (int)y0; ix0[wp] = (int)x0;
            cfy[wp] = yn - y0; cfx[wp] = xn - x0;
        }
        // --- stage strip weights for this k, pair-interleaved: [cpair][o] ---
        for (int i = tid; i < 32 * O_; i += 256) {
            int cp = i >> 6, o = i & 63;
            v2f t;
            t.x = wstrip[(o * C_ + 2 * cp)     * K_ + k];
            t.y = wstrip[(o * C_ + 2 * cp + 1) * K_ + k];
            wlds2[cp * WR2 + o] = t;
        }
        __syncthreads();

        // --- bilinear gather, two channels per iteration (shared coords) ---
        for (int i = tid; i < 32 * W_; i += 256) {
            int cp = i >> 7, p = i & 127;
            int y0 = iy0[p], x0 = ix0[p];
            int y1 = min(y0 + 1, H_ - 1), x1 = min(x0 + 1, W_ - 1);
            float fy = cfy[p], fx = cfx[p];
            const float* xc0 = xb + (2 * cp) * HW_;
            const float* xc1 = xc0 + HW_;
            int i00 = y0 * W_ + x0, i01 = y0 * W_ + x1;
            int i10 = y1 * W_ + x0, i11 = y1 * W_ + x1;
            float a00 = xc0[i00], a01 = xc0[i01], a10 = xc0[i10], a11 = xc0[i11];
            float b00 = xc1[i00], b01 = xc1[i01], b10 = xc1[i10], b11 = xc1[i11];
            float at = a00 + fx * (a01 - a00);
            float ab = a10 + fx * (a11 - a10);
            float bt = b00 + fx * (b01 - b00);
            float bb = b10 + fx * (b11 - b10);
            v2f t;
            t.x = at + fy * (ab - at);
            t.y = bt + fy * (bb - bt);
            slds2[cp * SR2 + p] = t;
        }
        __syncthreads();

        // --- accumulate: 16 chunks of c (WMMA K=4) x 4 output-channel tiles ---
        int pbase = wv * 16;
        #pragma unroll 4
        for (int cc = 0; cc < 16; ++cc) {
            int cpair = 2 * cc + half;               // halves carry K {0,1}/{2,3}
            v2f bfrag = slds2[cpair * SR2 + pbase + l16];   // one ds_load_b64
            #pragma unroll
            for (int mt = 0; mt < 4; ++mt) {
                v2f afrag = wlds2[cpair * WR2 + mt * 16 + l16];
                acc[mt] = __builtin_amdgcn_wmma_f32_16x16x4_f32(
                    false, afrag, false, bfrag, (short)0, acc[mt], false, false);
            }
        }
        __syncthreads();
    }

    // --- epilogue: C/D layout -> (o, pixel), add bias, store pre-GN output ---
    #pragma unroll
    for (int mt = 0; mt < 4; ++mt) {
        #pragma unroll
        for (int r = 0; r < 8; ++r) {
            int o  = mt * 16 + r + 8 * half;
            int wp = wv * 16 + l16;
            outraw[((long)(b * O_ + o)) * HW_ + h * W_ + wp] = acc[mt][r] + bstrip[o];
        }
    }
}

// ---------------------------------------------------------------------------
// Final GN(16 groups) + ReLU -> d_out
// ---------------------------------------------------------------------------
__global__ void out_apply_k(const float* __restrict__ raw, const float* __restrict__ stats,
                            const float* __restrict__ gamma, const float* __restrict__ beta,
                            float* __restrict__ dst) {
    int idx = blockIdx.x * blockDim.x + threadIdx.x;
    if (idx >= B_ * O_ * HW_) return;
    int ch = (idx / HW_) % O_;
    int b  = idx / (O_ * HW_);
    int g  = ch >> 2;                      // 16 groups of 4 channels
    float mu = stats[(b * 16 + g) * 2];
    float rs = stats[(b * 16 + g) * 2 + 1];
    float v = (raw[idx] - mu) * rs * gamma[ch] + beta[ch];
    dst[idx] = fmaxf(v, 0.f);
}

// ---------------------------------------------------------------------------
extern "C" void kernel_launch(void* const* d_in, const int* in_sizes, int n_in,
                              void* d_out, int out_size, void* d_ws, size_t ws_size,
                              hipStream_t stream) {
    const float* x        = (const float*)d_in[0];
    const float* w_off    = (const float*)d_in[1];
    const float* b_off    = (const float*)d_in[2];
    const float* w_theta  = (const float*)d_in[3];
    const float* b_theta  = (const float*)d_in[4];
    const float* gn_off_g = (const float*)d_in[5];
    const float* gn_off_b = (const float*)d_in[6];
    const float* gn_th_g  = (const float*)d_in[7];
    const float* gn_th_b  = (const float*)d_in[8];
    const float* w_strip  = (const float*)d_in[9];
    const float* b_strip  = (const float*)d_in[10];
    const float* gn_out_g = (const float*)d_in[11];
    const float* gn_out_b = (const float*)d_in[12];
    float* out = (float*)d_out;

    // workspace carve-up (floats)
    float* ws = (float*)d_ws;
    float* off_buf   = ws;                              // B*K*HW = 589824
    float* th_buf    = off_buf + (long)B_ * K_ * HW_;   // 589824
    float* outraw    = th_buf  + (long)B_ * K_ * HW_;   // B*O*HW = 4194304
    float* stats_off = outraw  + (long)B_ * O_ * HW_;   // 24
    float* stats_th  = stats_off + 32;                  // 24
    float* stats_out = stats_th  + 32;                  // 128

    const int nconv = B_ * K_ * HW_;
    const int nout  = B_ * O_ * HW_;

    // offset branch: conv3x3 -> GN(3) -> tanh
    conv_k<3><<<B_ * H_, 128, 0, stream>>>(x, w_off, b_off, off_buf);
    gn_stats_k<<<B_ * 3, 256, 0, stream>>>(off_buf, stats_off, 3, 3);
    gn_apply_k<<<(nconv + 255) / 256, 256, 0, stream>>>(off_buf, stats_off,
                                                        gn_off_g, gn_off_b, K_, 3, 0);
    // theta branch: conv5x5 -> GN(3) (atan folded into sampler algebra)
    conv_k<5><<<B_ * H_, 128, 0, stream>>>(x, w_theta, b_theta, th_buf);
    gn_stats_k<<<B_ * 3, 256, 0, stream>>>(th_buf, stats_th, 3, 3);
    gn_apply_k<<<(nconv + 255) / 256, 256, 0, stream>>>(th_buf, stats_th,
                                                        gn_th_g, gn_th_b, K_, 3, 1);
    // fused deform-sample + strip GEMM (WMMA f32)
    strip_wmma_k<<<B_ * H_, 256, 0, stream>>>(x, off_buf, th_buf, w_strip, b_strip, outraw);
    // final GN(16) + ReLU
    gn_stats_k<<<B_ * 16, 256, 0, stream>>>(outraw, stats_out, 4, 16);
    out_apply_k<<<(nout + 255) / 256, 256, 0, stream>>>(outraw, stats_out,
                                                        gn_out_g, gn_out_b, out);
}